// GINLayer_5901285065185
// MI455X (gfx1250) — compile-verified
//
#include <hip/hip_runtime.h>
#include <hip/hip_bf16.h>

#define NN 50000
#define DD 128
#define EE 1600000
#define NTILES (NN / 16)      // 3125 exact
#define BN_EPS 1e-5f
#define SLOPE 0.01f

typedef float v2f __attribute__((ext_vector_type(2)));
typedef float v8f __attribute__((ext_vector_type(8)));

#define ZS 132                // z / y1 tile row stride (128 cols + 4 pad)

// ---------------------------------------------------------------------------
// K0: zero agg buffer + global BN accumulators
// ---------------------------------------------------------------------------
__global__ void gin_zero_kernel(float* __restrict__ agg, float* __restrict__ stats) {
    int i = blockIdx.x * blockDim.x + threadIdx.x;
    int stride = gridDim.x * blockDim.x;
    int n4 = (NN * DD) / 4;
    float4 z4 = make_float4(0.f, 0.f, 0.f, 0.f);
    for (int j = i; j < n4; j += stride) ((float4*)agg)[j] = z4;
    if (i < 256) stats[i] = 0.f;
}

// ---------------------------------------------------------------------------
// K1: edge scatter-add. One wave per edge; each lane handles 4 features.
// h (25.6MB) + agg (25.6MB) both fit in the 192MB L2, so the fp32 atomics
// stay on-chip; HBM sees mostly the 12.8MB of indices.
// ---------------------------------------------------------------------------
__global__ void gin_scatter_kernel(const float* __restrict__ h,
                                   const int* __restrict__ src,
                                   const int* __restrict__ dst,
                                   float* __restrict__ agg) {
    int wid  = (blockIdx.x * blockDim.x + threadIdx.x) >> 5;
    int lane = threadIdx.x & 31;
    int nwaves = (gridDim.x * blockDim.x) >> 5;
    for (int e = wid; e < EE; e += nwaves) {
        int s = src[e];
        int d = dst[e];
        float4 v = ((const float4*)(h + (size_t)s * DD))[lane];
        float* p = agg + (size_t)d * DD + lane * 4;
        unsafeAtomicAdd(p + 0, v.x);
        unsafeAtomicAdd(p + 1, v.y);
        unsafeAtomicAdd(p + 2, v.z);
        unsafeAtomicAdd(p + 3, v.w);
    }
}

// ---------------------------------------------------------------------------
// K2: fused z = (1+eps)h + agg -> relu(z@W1+b1)@W2+b2 -> y, BN partial sums.
// 256 threads = 8 waves = 2 waves/SIMD. amdgpu_waves_per_eu(2) raises the
// VGPR cap to 512/wave so both weight-fragment sets (128 VGPRs) stay
// register-resident (no scratch spills). Each wave owns one 16-col output
// tile; 64x v_wmma_f32_16x16x4_f32 per node tile, fed by ds_load_2addr_b64
// A-fragments only. Next tile's h/agg prefetched during the WMMA work.
// ---------------------------------------------------------------------------
__global__ void __launch_bounds__(256)
__attribute__((amdgpu_waves_per_eu(2)))
gin_mlp_kernel(const float* __restrict__ h,
               const float* __restrict__ agg,
               const float* __restrict__ eps_p,
               const float* __restrict__ W1,
               const float* __restrict__ b1,
               const float* __restrict__ W2,
               const float* __restrict__ b2,
               float* __restrict__ y,
               float* __restrict__ gsum,
               float* __restrict__ gsq) {
    __shared__ float zT[16 * ZS];
    __shared__ float y1T[16 * ZS];
    __shared__ float ssum[DD];
    __shared__ float ssq[DD];

    int tid  = threadIdx.x;
    int lane = tid & 31;
    int half = lane >> 4;        // A/B fragment half: K pair offset
    int m    = lane & 15;        // row (A) / col-within-tile (B,C,D)
    int n0   = (tid >> 5) * 16;  // this wave's output column tile

    if (tid < DD) { ssum[tid] = 0.f; ssq[tid] = 0.f; }

    // Register-resident B fragments for W1 and W2 (loop-invariant per wave).
    v2f b1f[32], b2f[32];
    #pragma unroll
    for (int kk = 0; kk < 32; ++kk) {
        int ka = 4 * kk + 2 * half;
        b1f[kk].x = W1[ka * DD + n0 + m];
        b1f[kk].y = W1[(ka + 1) * DD + n0 + m];
        b2f[kk].x = W2[ka * DD + n0 + m];
        b2f[kk].y = W2[(ka + 1) * DD + n0 + m];
    }

    float epsv = 1.0f + eps_p[0];
    float bv1 = b1[n0 + m];
    float bv2 = b2[n0 + m];

    // z-staging coordinates: each thread owns 8 consecutive floats of one row
    int r    = tid >> 4;          // 0..15
    int ccol = (tid & 15) * 8;    // 0..120

    int tile = blockIdx.x;
    float4 h0, h1, a0, a1;
    if (tile < NTILES) {
        const float4* hp = (const float4*)(h   + (size_t)(tile * 16 + r) * DD + ccol);
        const float4* ap = (const float4*)(agg + (size_t)(tile * 16 + r) * DD + ccol);
        h0 = hp[0]; h1 = hp[1]; a0 = ap[0]; a1 = ap[1];
    }

    for (; tile < NTILES; tile += gridDim.x) {
        int row0 = tile * 16;

        // write z = (1+eps)*h + agg into LDS (16 x 128)
        {
            float* zp = zT + r * ZS + ccol;
            zp[0] = epsv * h0.x + a0.x;  zp[1] = epsv * h0.y + a0.y;
            zp[2] = epsv * h0.z + a0.z;  zp[3] = epsv * h0.w + a0.w;
            zp[4] = epsv * h1.x + a1.x;  zp[5] = epsv * h1.y + a1.y;
            zp[6] = epsv * h1.z + a1.z;  zp[7] = epsv * h1.w + a1.w;
        }
        __syncthreads();

        // prefetch next tile's h/agg while the WMMAs run
        int nt = tile + gridDim.x;
        if (nt < NTILES) {
            const float4* hp = (const float4*)(h   + (size_t)(nt * 16 + r) * DD + ccol);
            const float4* ap = (const float4*)(agg + (size_t)(nt * 16 + r) * DD + ccol);
            h0 = hp[0]; h1 = hp[1]; a0 = ap[0]; a1 = ap[1];
        }

        // GEMM1: c1 = z @ W1 (K=128 via 32 WMMA steps, B in registers)
        v8f c1 = {0.f, 0.f, 0.f, 0.f, 0.f, 0.f, 0.f, 0.f};
        #pragma unroll
        for (int kk = 0; kk < 32; ++kk) {
            int ka = 4 * kk + 2 * half;
            v2f a;
            a.x = zT[m * ZS + ka];
            a.y = zT[m * ZS + ka + 1];
            c1 = __builtin_amdgcn_wmma_f32_16x16x4_f32(false, a, false, b1f[kk],
                                                       (short)0, c1, false, false);
        }
        // bias + relu, write intermediate tile to LDS (C/D: VGPR i -> row i + 8*half)
        #pragma unroll
        for (int i = 0; i < 8; ++i) {
            float v = c1[i] + bv1;
            v = v > 0.f ? v : 0.f;
            y1T[(i + 8 * half) * ZS + n0 + m] = v;
        }
        __syncthreads();

        // GEMM2: c2 = relu(...) @ W2
        v8f c2 = {0.f, 0.f, 0.f, 0.f, 0.f, 0.f, 0.f, 0.f};
        #pragma unroll
        for (int kk = 0; kk < 32; ++kk) {
            int ka = 4 * kk + 2 * half;
            v2f a;
            a.x = y1T[m * ZS + ka];
            a.y = y1T[m * ZS + ka + 1];
            c2 = __builtin_amdgcn_wmma_f32_16x16x4_f32(false, a, false, b2f[kk],
                                                       (short)0, c2, false, false);
        }
        // bias, store y, accumulate BN partials (all 8 values share column n0+m)
        float s = 0.f, sq = 0.f;
        #pragma unroll
        for (int i = 0; i < 8; ++i) {
            float v = c2[i] + bv2;
            y[(size_t)(row0 + i + 8 * half) * DD + n0 + m] = v;
            s  += v;
            sq += v * v;
        }
        atomicAdd(&ssum[n0 + m], s);
        atomicAdd(&ssq[n0 + m], sq);
        __syncthreads();   // guards zT/y1T reuse next iteration
    }

    // flush block-local stats (2 global fp atomics per column per block)
    if (tid < DD) {
        unsafeAtomicAdd(&gsum[tid], ssum[tid]);
        unsafeAtomicAdd(&gsq[tid],  ssq[tid]);
    }
}

// ---------------------------------------------------------------------------
// K3: finalize BN stats -> mean, invstd
// ---------------------------------------------------------------------------
__global__ void gin_stats_kernel(const float* __restrict__ gsum,
                                 const float* __restrict__ gsq,
                                 float* __restrict__ mean,
                                 float* __restrict__ invstd) {
    int c = threadIdx.x;
    if (c < DD) {
        float inv_n = 1.0f / (float)NN;
        float mu  = gsum[c] * inv_n;
        float var = gsq[c] * inv_n - mu * mu;
        mean[c]   = mu;
        invstd[c] = rsqrtf(var + BN_EPS);
    }
}

// ---------------------------------------------------------------------------
// K4: out = h + leaky_relu(gamma * (y - mean) * invstd + beta)
// ---------------------------------------------------------------------------
__global__ void gin_finalize_kernel(const float* __restrict__ h,
                                    const float* __restrict__ y,
                                    const float* __restrict__ mean,
                                    const float* __restrict__ invstd,
                                    const float* __restrict__ gamma,
                                    const float* __restrict__ beta,
                                    float* __restrict__ out) {
    int idx = blockIdx.x * blockDim.x + threadIdx.x;
    int stride = gridDim.x * blockDim.x;
    int total = (NN * DD) / 4;
    for (int i = idx; i < total; i += stride) {
        int c4 = i & 31;  // 32 float4 per row of 128
        float4 yv = ((const float4*)y)[i];
        float4 hv = ((const float4*)h)[i];
        float4 mu = ((const float4*)mean)[c4];
        float4 is = ((const float4*)invstd)[c4];
        float4 g  = ((const float4*)gamma)[c4];
        float4 b  = ((const float4*)beta)[c4];
        float4 o;
        float t;
        t = (yv.x - mu.x) * is.x * g.x + b.x; o.x = hv.x + (t >= 0.f ? t : SLOPE * t);
        t = (yv.y - mu.y) * is.y * g.y + b.y; o.y = hv.y + (t >= 0.f ? t : SLOPE * t);
        t = (yv.z - mu.z) * is.z * g.z + b.z; o.z = hv.z + (t >= 0.f ? t : SLOPE * t);
        t = (yv.w - mu.w) * is.w * g.w + b.w; o.w = hv.w + (t >= 0.f ? t : SLOPE * t);
        ((float4*)out)[i] = o;
    }
}

// ---------------------------------------------------------------------------
extern "C" void kernel_launch(void* const* d_in, const int* in_sizes, int n_in,
                              void* d_out, int out_size, void* d_ws, size_t ws_size,
                              hipStream_t stream) {
    const float* h     = (const float*)d_in[0];
    const int*   src   = (const int*)d_in[1];
    const int*   dst   = (const int*)d_in[2];
    const float* eps_p = (const float*)d_in[3];
    const float* W1    = (const float*)d_in[4];
    const float* b1    = (const float*)d_in[5];
    const float* W2    = (const float*)d_in[6];
    const float* b2    = (const float*)d_in[7];
    const float* gamma = (const float*)d_in[8];
    const float* beta  = (const float*)d_in[9];
    float* out = (float*)d_out;

    // workspace layout (floats)
    float* ws     = (float*)d_ws;
    float* agg    = ws;                        // N*D
    float* ybuf   = agg + (size_t)NN * DD;     // N*D
    float* gsum   = ybuf + (size_t)NN * DD;    // 128
    float* gsq    = gsum + DD;                 // 128
    float* meanb  = gsq + DD;                  // 128
    float* invstd = meanb + DD;                // 128

    // K0: zero agg + stats accumulators
    gin_zero_kernel<<<4096, 256, 0, stream>>>(agg, gsum);

    // K1: edge scatter-add (one wave per edge)
    gin_scatter_kernel<<<4096, 256, 0, stream>>>(h, src, dst, agg);

    // K2: fused GIN MLP with fp32 WMMA + BN partial sums
    gin_mlp_kernel<<<512, 256, 0, stream>>>(h, agg, eps_p, W1, b1, W2, b2,
                                            ybuf, gsum, gsq);

    // K3: finalize mean / invstd
    gin_stats_kernel<<<1, 128, 0, stream>>>(gsum, gsq, meanb, invstd);

    // K4: BN apply + leaky relu + residual
    gin_finalize_kernel<<<2048, 256, 0, stream>>>(h, ybuf, meanb, invstd, gamma, beta, out);
}